// TetRenderer_19241453486106
// MI455X (gfx1250) — compile-verified
//
#include <hip/hip_runtime.h>
#include <hip/hip_bf16.h>
#include <stdint.h>

#define HH 96
#define WW 96
#define NPIX (HH * WW)
#define NV 512
#define NF 1024
#define EPSF 1e-8f

typedef float v2f __attribute__((ext_vector_type(2)));
typedef float v8f __attribute__((ext_vector_type(8)));
typedef int b128_t __attribute__((vector_size(16)));  // async-LDS builtin arg type

#define __AS1 __attribute__((address_space(1)))
#define __AS3 __attribute__((address_space(3)))

#if defined(__has_builtin)
#if __has_builtin(__builtin_amdgcn_global_load_async_to_lds_b128) && \
    __has_builtin(__builtin_amdgcn_s_wait_asynccnt)
#define USE_ASYNC_LDS 1
#endif
#endif

// -------------------------------------------------------------------------
// Kernel 1: vertex transform via V_WMMA_F32_16X16X4_F32, software-pipelined.
// D(16x16) = A(16x4 verts-homog) x B(4x16 = P^T zero-padded) per 16-vert chunk.
// One wave (32 lanes) processes all 512 verts in 32 WMMAs; the next chunk's
// A-operand load is issued before the current WMMA so load latency overlaps
// the matrix op instead of serializing on s_wait_loadcnt.
// A layout (f32 16x4): lane%16 = M (vert), lane-half+vgpr = K (coord).
// B layout (f32 4x16): lane%16 = N (out coord), lane-half+vgpr = K.
// D layout: lane%16 = N, vgpr r -> M = r + 8*(lane>=16).
// -------------------------------------------------------------------------
__global__ __launch_bounds__(32) void xform_wmma(
    const float* __restrict__ verts, const float* __restrict__ mv,
    const float* __restrict__ proj, float* __restrict__ clip) {
  const int lane = threadIdx.x & 31;
  // P = proj @ mv (4x4, uniform per lane)
  float P[16];
#pragma unroll
  for (int i = 0; i < 4; ++i)
#pragma unroll
    for (int j = 0; j < 4; ++j) {
      float s = 0.f;
#pragma unroll
      for (int k = 0; k < 4; ++k) s = fmaf(proj[i * 4 + k], mv[k * 4 + j], s);
      P[i * 4 + j] = s;
    }
  const int m = lane & 15;
  const bool hi = lane >= 16;
  // B[k][n] = P[n][k] for n<4, else 0
  v2f b;
  b.x = (m < 4) ? P[m * 4 + (hi ? 2 : 0)] : 0.0f;
  b.y = (m < 4) ? P[m * 4 + (hi ? 3 : 1)] : 0.0f;

  auto loadA = [&](int chunk) -> v2f {
    const int v = chunk * 16 + m;
    const float vx = verts[v * 3 + 0];
    const float vy = verts[v * 3 + 1];
    const float vz = verts[v * 3 + 2];
    v2f a;
    a.x = hi ? vz : vx;   // K=2 : K=0
    a.y = hi ? 1.0f : vy; // K=3 : K=1  (homogeneous w=1)
    return a;
  };

  v2f a_cur = loadA(0);
  for (int chunk = 0; chunk < NV / 16; ++chunk) {
    const int nc = (chunk + 1 < NV / 16) ? (chunk + 1) : chunk;
    v2f a_next = loadA(nc);  // issue next load before WMMA
    v8f c = {0.f, 0.f, 0.f, 0.f, 0.f, 0.f, 0.f, 0.f};
    c = __builtin_amdgcn_wmma_f32_16x16x4_f32(false, a_cur, false, b, (short)0,
                                              c, false, false);
    if (m < 4) {  // lanes holding valid output coords N=0..3
      const int vb = chunk * 16 + (hi ? 8 : 0);
#pragma unroll
      for (int r = 0; r < 8; ++r) clip[(vb + r) * 4 + m] = c[r];
    }
    a_cur = a_next;
  }
}

// -------------------------------------------------------------------------
// Kernel 2: per-face setup -> planar edge/depth/color functions of (px,py).
// fP[f]:  {A0,B0,C0,da}{A1,B1,C1,db}{A2,B2,C2,dc}   (phase-1 data, LDS-cached)
// fAlpha[f]: clipped opacity, 0 if face invalid (degenerate / behind camera)
// fC[f]:  {pa_r,pb_r,pc_r,alpha}{pa_g,pb_g,pc_g,0}{pa_b,pb_b,pc_b,0}
// -------------------------------------------------------------------------
__global__ __launch_bounds__(256) void face_setup(
    const int* __restrict__ faces, const float* __restrict__ verts_color,
    const float* __restrict__ opac, const float* __restrict__ intense,
    const float* __restrict__ clip, float4* __restrict__ fP,
    float* __restrict__ fAlpha, float4* __restrict__ fC) {
  const int f = blockIdx.x * blockDim.x + threadIdx.x;
  if (f >= NF) return;
  const int idx[3] = {faces[f * 3 + 0], faces[f * 3 + 1], faces[f * 3 + 2]};
  float X[3], Y[3], Z[3];
  bool front = true;
#pragma unroll
  for (int k = 0; k < 3; ++k) {
    const float* cp = clip + idx[k] * 4;
    const float cw = cp[3];
    front = front && (cw > EPSF);
    const float iw = 1.0f / cw;
    X[k] = (cp[0] * iw + 1.0f) * 0.5f * (float)WW;
    Y[k] = (1.0f - cp[1] * iw) * 0.5f * (float)HH;
    Z[k] = cp[2] * iw;
  }
  const float area =
      (X[2] - X[0]) * (Y[1] - Y[0]) - (Y[2] - Y[0]) * (X[1] - X[0]);
  const bool valid = front && (fabsf(area) > EPSF);
  const float area_s = (fabsf(area) < EPSF) ? 1.0f : area;
  const float inv = 1.0f / area_s;
  // edge planes: w_i(px,py) = A_i*px + B_i*py + C_i
  const float A0 = Y[2] - Y[1], B0 = -(X[2] - X[1]);
  const float C0 = -(X[1] * A0 + Y[1] * B0);
  const float A1 = Y[0] - Y[2], B1 = -(X[0] - X[2]);
  const float C1 = -(X[2] * A1 + Y[2] * B1);
  const float A2 = Y[1] - Y[0], B2 = -(X[1] - X[0]);
  const float C2 = -(X[0] * A2 + Y[0] * B2);
  // depth plane = (w0*z0 + w1*z1 + w2*z2) / area_s
  const float da = (A0 * Z[0] + A1 * Z[1] + A2 * Z[2]) * inv;
  const float db = (B0 * Z[0] + B1 * Z[1] + B2 * Z[2]) * inv;
  const float dc = (C0 * Z[0] + C1 * Z[1] + C2 * Z[2]) * inv;
  float o = opac[f];
  o = fminf(fmaxf(o, 0.0f), 0.999f);
  const float alpha = valid ? o : 0.0f;
  const float it = intense[f];
  float pa[3], pb[3], pc[3];
#pragma unroll
  for (int c = 0; c < 3; ++c) {
    const float c0 = verts_color[idx[0] * 3 + c];
    const float c1 = verts_color[idx[1] * 3 + c];
    const float c2 = verts_color[idx[2] * 3 + c];
    pa[c] = (A0 * c0 + A1 * c1 + A2 * c2) * inv * it;
    pb[c] = (B0 * c0 + B1 * c1 + B2 * c2) * inv * it;
    pc[c] = (C0 * c0 + C1 * c1 + C2 * c2) * inv * it;
  }
  fP[f * 3 + 0] = make_float4(A0, B0, C0, da);
  fP[f * 3 + 1] = make_float4(A1, B1, C1, db);
  fP[f * 3 + 2] = make_float4(A2, B2, C2, dc);
  fAlpha[f] = alpha;
  fC[f * 3 + 0] = make_float4(pa[0], pb[0], pc[0], alpha);
  fC[f * 3 + 1] = make_float4(pa[1], pb[1], pc[1], 0.0f);
  fC[f * 3 + 2] = make_float4(pa[2], pb[2], pc[2], 0.0f);
}

// -------------------------------------------------------------------------
// Kernel 3: one thread per pixel. Stage the 52KB face table into LDS via the
// CDNA5 async global->LDS DMA path (ASYNCcnt) when available. Phase 1:
// stream all faces from LDS, build sorted (depth,faceIdx) keys (monotone
// float->u32 pack, stable on index). Phase 2: front-to-back composite.
// -------------------------------------------------------------------------
__global__ __launch_bounds__(256) void raster(
    const float4* __restrict__ fP, const float* __restrict__ fAlpha,
    const float4* __restrict__ fC, const float* __restrict__ bg,
    float* __restrict__ outc, float* __restrict__ outd,
    float* __restrict__ outa) {
  __shared__ float4 sP[NF * 3];  // 48 KB of 320 KB/WGP LDS
  __shared__ float sA[NF];       //  4 KB

#if defined(USE_ASYNC_LDS)
  {
    // global_load_async_to_lds_b128: copies 16 bytes from the global pointer to the LDS pointer, tracked by ASYNCcnt
    const __AS1 char* gp = (const __AS1 char*)(const char*)fP;
    __AS3 char* lp = (__AS3 char*)(char*)sP;
    for (int i = threadIdx.x; i < NF * 3; i += blockDim.x)
      __builtin_amdgcn_global_load_async_to_lds_b128(
          (__AS1 b128_t*)(gp + 16u * i), (__AS3 b128_t*)(lp + 16u * i), 0, 0);
    const __AS1 char* ga = (const __AS1 char*)(const char*)fAlpha;
    __AS3 char* la = (__AS3 char*)(char*)sA;
    for (int i = threadIdx.x; i < NF / 4; i += blockDim.x)
      __builtin_amdgcn_global_load_async_to_lds_b128(
          (__AS1 b128_t*)(ga + 16u * i), (__AS3 b128_t*)(la + 16u * i), 0, 0);
    __builtin_amdgcn_s_wait_asynccnt(0);
  }
#else
  for (int i = threadIdx.x; i < NF * 3; i += blockDim.x) sP[i] = fP[i];
  for (int i = threadIdx.x; i < NF; i += blockDim.x) sA[i] = fAlpha[i];
#endif
  __syncthreads();

  // warm WGP$/L2 with the phase-2 color-plane table while phase 1 runs
  __builtin_prefetch(&fC[(threadIdx.x * 12) % (NF * 3)], 0, 0);

  const int p = blockIdx.x * blockDim.x + threadIdx.x;
  if (p >= NPIX) return;
  const float px = (float)(p % WW) + 0.5f;
  const float py = (float)(p / WW) + 0.5f;

  unsigned long long keys[NF];  // scratch; expected occupancy ~60 entries
  int K = 0;
  for (int f = 0; f < NF; ++f) {
    const float al = sA[f];
    const float4 e0 = sP[f * 3 + 0];
    const float4 e1 = sP[f * 3 + 1];
    const float4 e2 = sP[f * 3 + 2];
    const float w0 = fmaf(e0.x, px, fmaf(e0.y, py, e0.z));
    const float w1 = fmaf(e1.x, px, fmaf(e1.y, py, e1.z));
    const float w2 = fmaf(e2.x, px, fmaf(e2.y, py, e2.z));
    const bool pos = (w0 >= 0.f) && (w1 >= 0.f) && (w2 >= 0.f);
    const bool neg = (w0 <= 0.f) && (w1 <= 0.f) && (w2 <= 0.f);
    if ((pos || neg) && (al > 0.f)) {
      const float d = fmaf(e0.w, px, fmaf(e1.w, py, e2.w));
      unsigned u = __float_as_uint(d);
      u = (u & 0x80000000u) ? ~u : (u | 0x80000000u);  // monotone pack
      keys[K++] = ((unsigned long long)u << 32) | (unsigned)f;
    }
  }
  // insertion sort (stable on face index via low 32 bits)
  for (int i = 1; i < K; ++i) {
    const unsigned long long kk = keys[i];
    int j = i - 1;
    while (j >= 0 && keys[j] > kk) {
      keys[j + 1] = keys[j];
      --j;
    }
    keys[j + 1] = kk;
  }
  // front-to-back composite
  float t = 1.0f, cr = 0.f, cg = 0.f, cb = 0.f, dd = 0.f;
  for (int i = 0; i < K; ++i) {
    const unsigned long long kk = keys[i];
    const int f = (int)(kk & 0xffffffffu);
    const unsigned m = (unsigned)(kk >> 32);
    const unsigned u = (m & 0x80000000u) ? (m ^ 0x80000000u) : ~m;
    const float d = __uint_as_float(u);
    const float4 c0 = fC[f * 3 + 0];
    const float4 c1 = fC[f * 3 + 1];
    const float4 c2 = fC[f * 3 + 2];
    const float al = c0.w;
    const float w = t * al;
    cr = fmaf(w, fmaf(c0.x, px, fmaf(c0.y, py, c0.z)), cr);
    cg = fmaf(w, fmaf(c1.x, px, fmaf(c1.y, py, c1.z)), cg);
    cb = fmaf(w, fmaf(c2.x, px, fmaf(c2.y, py, c2.z)), cb);
    dd = fmaf(w, d, dd);
    t *= (1.0f - al);
    if (t < 1e-7f) break;  // transmittance dead: error << fp32 noise
  }
  outc[p * 3 + 0] = fmaf(t, bg[0], cr);
  outc[p * 3 + 1] = fmaf(t, bg[1], cg);
  outc[p * 3 + 2] = fmaf(t, bg[2], cb);
  outd[p] = dd;
  outa[p] = ((1.0f - t) > 0.5f) ? 1.0f : 0.0f;
}

// -------------------------------------------------------------------------
extern "C" void kernel_launch(void* const* d_in, const int* in_sizes, int n_in,
                              void* d_out, int out_size, void* d_ws,
                              size_t ws_size, hipStream_t stream) {
  const float* verts = (const float*)d_in[0];       // (512,3)
  const int* faces = (const int*)d_in[1];           // (1024,3)
  const float* vcol = (const float*)d_in[2];        // (512,3)
  const float* fop = (const float*)d_in[3];         // (1024,)
  const float* mv = (const float*)d_in[4];          // (1,4,4)
  const float* proj = (const float*)d_in[5];        // (1,4,4)
  const float* fint = (const float*)d_in[7];        // (1,1024)
  const float* bg = (const float*)d_in[11];         // (3,)

  float* ws = (float*)d_ws;
  float* clip = ws;                                   // 2048 floats
  float4* fP = (float4*)(ws + 2048);                  // 3072 float4
  float* fAlpha = ws + 2048 + 12288;                  // 1024 floats
  float4* fC = (float4*)(ws + 2048 + 12288 + 1024);   // 3072 float4

  float* outc = (float*)d_out;          // (96,96,3)
  float* outd = outc + NPIX * 3;        // (96,96)
  float* outa = outd + NPIX;            // (96,96)

  xform_wmma<<<1, 32, 0, stream>>>(verts, mv, proj, clip);
  face_setup<<<NF / 256, 256, 0, stream>>>(faces, vcol, fop, fint, clip, fP,
                                           fAlpha, fC);
  raster<<<(NPIX + 255) / 256, 256, 0, stream>>>(fP, fAlpha, fC, bg, outc,
                                                 outd, outa);
}